// minibatch_discrimination_19670950216008
// MI455X (gfx1250) — compile-verified
//
#include <hip/hip_runtime.h>

// Problem constants from the reference
#define N_  256
#define A_  1024
#define B_  64
#define C_  16
#define BC_ (B_ * C_)   // 1024

typedef float v2f __attribute__((ext_vector_type(2)));
typedef float v8f __attribute__((ext_vector_type(8)));

// ---------------------------------------------------------------------------
// Stage 1: mat[256,1024] = x[256,1024] @ weight[1024,1024] in pure f32 WMMA.
// One wave32 per 16x16 output tile; K-loop of 256 x V_WMMA_F32_16X16X4_F32.
//
// A-operand layout (16x4 f32, ISA 7.12.2): lane L (row = L&15) holds
//   VGPR0 = A[row, k + 2*(L>>4)], VGPR1 = A[row, k + 2*(L>>4) + 1]
//   -> one contiguous float2 load per lane.
// B-operand layout (4x16 f32): lane L (col = L&15) holds
//   VGPR0 = B[k + 2*(L>>4), col], VGPR1 = B[k + 2*(L>>4) + 1, col]
//   -> two b32 loads strided by the row pitch.
// D layout: VGPR r -> M = tileM + r + 8*(L>>4), N = tileN + (L&15).
// ---------------------------------------------------------------------------
__global__ __launch_bounds__(256) void gemm_f32_wmma(
    const float* __restrict__ x, const float* __restrict__ w,
    float* __restrict__ mat) {
  const int wave  = (blockIdx.x * 256 + threadIdx.x) >> 5;  // 0..1023
  const int lane  = threadIdx.x & 31;
  const int tileN = (wave & 63) << 4;   // 64 tiles along N (1024/16)
  const int tileM = (wave >> 6) << 4;   // 16 tiles along M (256/16)
  const int l15   = lane & 15;
  const int koff  = (lane >> 4) << 1;   // 0 or 2

  const float* __restrict__ aptr = x + (size_t)(tileM + l15) * A_ + koff;
  const float* __restrict__ bptr = w + (size_t)koff * BC_ + tileN + l15;

  v8f acc = (v8f)0.0f;
#pragma unroll 4
  for (int k = 0; k < A_; k += 4) {
    v2f a = *(const v2f*)(aptr + k);   // A[row, k+koff .. k+koff+1]
    v2f b;
    b.x = bptr[(size_t)k * BC_];           // B[k+koff,   col]
    b.y = bptr[(size_t)k * BC_ + BC_];     // B[k+koff+1, col]
    // 8 args: (neg_a, A, neg_b, B, c_mod, C, reuse_a, reuse_b)
    acc = __builtin_amdgcn_wmma_f32_16x16x4_f32(
        false, a, false, b, (short)0, acc, false, false);
  }

  const int mbase = tileM + ((lane >> 4) << 3);  // +8 for upper half-wave
  float* __restrict__ outp = mat + (size_t)mbase * BC_ + tileN + l15;
#pragma unroll
  for (int r = 0; r < 8; ++r) outp[(size_t)r * BC_] = acc[r];
}

// ---------------------------------------------------------------------------
// Stage 2: out[j,b] = bias[b] + sum_i exp(-sum_c |mat[i,b,c]-mat[j,b,c]|)
// One block per b. Stage mat[:,b,:] (256x16 f32 = 16 KB) into LDS; thread j
// keeps its own row in registers from the staging load; the i-sweep reads LDS
// with all lanes at the same address (broadcast, conflict-free).
// ---------------------------------------------------------------------------
__global__ __launch_bounds__(256) void pairwise_l1_exp(
    const float* __restrict__ mat, const float* __restrict__ bias,
    float* __restrict__ out) {
  __shared__ float lds[N_ * C_];
  const int b = blockIdx.x;    // 0..63
  const int j = threadIdx.x;   // 0..255

  const float4* __restrict__ src =
      (const float4*)(mat + (size_t)j * BC_ + (size_t)b * C_);
  float4 r0 = src[0], r1 = src[1], r2 = src[2], r3 = src[3];
  float4* dst = (float4*)(lds + j * C_);
  dst[0] = r0; dst[1] = r1; dst[2] = r2; dst[3] = r3;
  __syncthreads();

  float mj[16] = {r0.x, r0.y, r0.z, r0.w, r1.x, r1.y, r1.z, r1.w,
                  r2.x, r2.y, r2.z, r2.w, r3.x, r3.y, r3.z, r3.w};

  const float kNegLog2e = -1.44269504088896340736f;
  float sum = 0.0f;
#pragma unroll 2
  for (int i = 0; i < N_; ++i) {
    const float4* row = (const float4*)(lds + i * C_);
    float4 q0 = row[0], q1 = row[1], q2 = row[2], q3 = row[3];
    float l1 = fabsf(q0.x - mj[0])  + fabsf(q0.y - mj[1])
             + fabsf(q0.z - mj[2])  + fabsf(q0.w - mj[3])
             + fabsf(q1.x - mj[4])  + fabsf(q1.y - mj[5])
             + fabsf(q1.z - mj[6])  + fabsf(q1.w - mj[7])
             + fabsf(q2.x - mj[8])  + fabsf(q2.y - mj[9])
             + fabsf(q2.z - mj[10]) + fabsf(q2.w - mj[11])
             + fabsf(q3.x - mj[12]) + fabsf(q3.y - mj[13])
             + fabsf(q3.z - mj[14]) + fabsf(q3.w - mj[15]);
    sum += __builtin_amdgcn_exp2f(l1 * kNegLog2e);  // exp(-l1)
  }

  out[(size_t)j * B_ + b] = sum + bias[b];
}

extern "C" void kernel_launch(void* const* d_in, const int* in_sizes, int n_in,
                              void* d_out, int out_size, void* d_ws, size_t ws_size,
                              hipStream_t stream) {
  const float* x    = (const float*)d_in[0];  // [256, 1024]
  const float* w    = (const float*)d_in[1];  // [1024, 1024]
  const float* bias = (const float*)d_in[2];  // [64]
  float* mat = (float*)d_ws;                  // [256, 1024] scratch (1 MB)
  float* out = (float*)d_out;                 // [256, 64]

  (void)in_sizes; (void)n_in; (void)out_size; (void)ws_size;

  // 1024 waves = 128 blocks x 8 waves; one 16x16 tile per wave.
  gemm_f32_wmma<<<dim3(128), dim3(256), 0, stream>>>(x, w, mat);
  // One block per b-slice.
  pairwise_l1_exp<<<dim3(B_), dim3(256), 0, stream>>>(mat, bias, out);
}